// Projection_76854144795318
// MI455X (gfx1250) — compile-verified
//
#include <hip/hip_runtime.h>
#include <math.h>
#include <stdint.h>

// Multi-resolution trilinear feature interpolation (NeRF-style grid lookup).
// Only features1/2/3 are used (BLOCK_IDS = 1,2,3). Output: [B, N, 32+64+128].
//
// Single fused dispatch; per level:
//  - each point handled by C/4 lanes; each lane gathers 8 float4 corners
//    (global_load_b128, coalesced across the point group) and emits one
//    float4 of output.
//  - coords staged into LDS with CDNA5 async-to-LDS loads
//    (global_load_async_to_lds_b32, ASYNCcnt) — no VGPR round trip.
//  - output written with non-temporal b128 stores so the 179MB write stream
//    does not evict the 84MB feature working set from the 192MB L2.

typedef float v4f __attribute__((ext_vector_type(4)));

// Stage nstage floats of coords into LDS via the async data mover.
// Issuing lanes are EXEC-masked; all lanes then drain ASYNCcnt and barrier.
__device__ __forceinline__ void stage_coords_async(
    const float* __restrict__ coords, float* sc,
    int base_pid, int npts, int t, int nstage)
{
    if (t < nstage) {
        const int g = base_pid * 3 + t;
        if (g < npts * 3) {
            const uint32_t lds_off = (uint32_t)(uintptr_t)(sc + t); // LDS byte addr = low 32 bits
            const uint32_t goff    = (uint32_t)g * 4u;              // byte offset from SGPR base
            asm volatile("global_load_async_to_lds_b32 %0, %1, %2"
                         :: "v"(lds_off), "v"(goff), "s"(coords)
                         : "memory");
        }
    }
    asm volatile("s_wait_asynccnt 0" ::: "memory");
    __syncthreads();
}

template <int DIM, int C, int COFF, int TOTC>
__device__ __forceinline__ void trilerp_block(
    const float* __restrict__ feat,     // [B, DIM, DIM, DIM, C]
    const float* __restrict__ coords,   // [B*N, 3] flat
    float* __restrict__ out,            // [B*N, TOTC] flat
    float* sc, int blk, int npts, int N)
{
    constexpr int TPP = C / 4;      // lanes per point
    constexpr int P   = 256 / TPP;  // points per block

    const int t        = threadIdx.x;
    const int base_pid = blk * P;

    stage_coords_async(coords, sc, base_pid, npts, t, P * 3);

    const int lp  = t / TPP;        // local point in block
    const int v   = t % TPP;        // float4 channel slot
    const int pid = base_pid + lp;
    if (pid >= npts) return;
    const int b = pid / N;

    // Reference: idx = clip(coord * (0.5^p * 128), 0.01, DIM - 1.01).
    // For levels 1..3, 0.5^p * 128 == DIM exactly.
    const float hi = (float)DIM - 1.01f;
    const float fx = fminf(fmaxf(sc[lp * 3 + 0] * (float)DIM, 0.01f), hi);
    const float fy = fminf(fmaxf(sc[lp * 3 + 1] * (float)DIM, 0.01f), hi);
    const float fz = fminf(fmaxf(sc[lp * 3 + 2] * (float)DIM, 0.01f), hi);

    const float x1 = floorf(fx), x2 = ceilf(fx);
    const float y1 = floorf(fy), y2 = ceilf(fy);
    const float z1 = floorf(fz), z2 = ceilf(fz);
    const float wx = fx - x1, wx2 = x2 - fx;
    const float wy = fy - y1, wy2 = y2 - fy;
    const float wz = fz - z1, wz2 = z2 - fz;

    const int ix1 = (int)x1, ix2 = (int)x2;
    const int iy1 = (int)y1, iy2 = (int)y2;
    const int iz1 = (int)z1, iz2 = (int)z2;

    const int sx  = DIM * DIM * C;
    const int sy  = DIM * C;
    const int ob  = b * DIM * sx + v * 4;     // batch + channel-slot offset
    const int ax1 = ix1 * sx, ax2 = ix2 * sx;
    const int ay1 = iy1 * sy, ay2 = iy2 * sy;
    const int az1 = iz1 * C,  az2 = iz2 * C;

    const v4f* __restrict__ f4 = reinterpret_cast<const v4f*>(feat);
    // z = z1 plane
    const v4f q11a = f4[(ob + ax1 + ay1 + az1) >> 2];
    const v4f q21a = f4[(ob + ax2 + ay1 + az1) >> 2];
    const v4f q12a = f4[(ob + ax1 + ay2 + az1) >> 2];
    const v4f q22a = f4[(ob + ax2 + ay2 + az1) >> 2];
    // z = z2 plane
    const v4f q11b = f4[(ob + ax1 + ay1 + az2) >> 2];
    const v4f q21b = f4[(ob + ax2 + ay1 + az2) >> 2];
    const v4f q12b = f4[(ob + ax1 + ay2 + az2) >> 2];
    const v4f q22b = f4[(ob + ax2 + ay2 + az2) >> 2];

    // Same association order as the reference:
    //   lerp_y = (q22*wx + q12*wx2)*wy + (q21*wx + q11*wx2)*wy2
    //   out    = lerp_y(z2)*wz + lerp_y(z1)*wz2
    const v4f la = (q22a * wx + q12a * wx2) * wy + (q21a * wx + q11a * wx2) * wy2;
    const v4f lb = (q22b * wx + q12b * wx2) * wy + (q21b * wx + q11b * wx2) * wy2;
    const v4f r  = lb * wz + la * wz2;

    // Non-temporal 128-bit store: output is write-once/never-read.
    v4f* __restrict__ o4 =
        reinterpret_cast<v4f*>(out) + ((pid * TOTC + COFF + v * 4) >> 2);
    __builtin_nontemporal_store(r, o4);
}

__global__ __launch_bounds__(256) void trilerp_fused(
    const float* __restrict__ f1, const float* __restrict__ f2,
    const float* __restrict__ f3, const float* __restrict__ coords,
    float* __restrict__ out, int npts, int N, int g1, int g2)
{
    __shared__ float sc[96];   // max P*3 = 32*3
    const int bid = blockIdx.x;
    if (bid < g1) {
        trilerp_block<64, 32, 0, 224>(f1, coords, out, sc, bid, npts, N);
    } else if (bid < g1 + g2) {
        trilerp_block<32, 64, 32, 224>(f2, coords, out, sc, bid - g1, npts, N);
    } else {
        trilerp_block<16, 128, 96, 224>(f3, coords, out, sc, bid - g1 - g2, npts, N);
    }
}

extern "C" void kernel_launch(void* const* d_in, const int* in_sizes, int n_in,
                              void* d_out, int out_size, void* d_ws, size_t ws_size,
                              hipStream_t stream) {
    // Input order: features0..features4, mesh_coords. Only 1,2,3 and coords used.
    const float* f1 = (const float*)d_in[1];   // [B,64,64,64,32]
    const float* f2 = (const float*)d_in[2];   // [B,32,32,32,64]
    const float* f3 = (const float*)d_in[3];   // [B,16,16,16,128]
    const float* mc = (const float*)d_in[5];   // [B,N,3]
    float* out = (float*)d_out;

    const int B    = in_sizes[1] / (64 * 64 * 64 * 32);
    const int N    = in_sizes[5] / (B * 3);
    const int npts = B * N;

    const int g1 = (npts + 31) / 32;   // level1: 32 points/block (8 lanes/point)
    const int g2 = (npts + 15) / 16;   // level2: 16 points/block (16 lanes/point)
    const int g3 = (npts + 7) / 8;     // level3:  8 points/block (one wave32/point)

    trilerp_fused<<<g1 + g2 + g3, 256, 0, stream>>>(f1, f2, f3, mc, out,
                                                    npts, N, g1, g2);
}